// ChildSumTreeLSTM_29154238005586
// MI455X (gfx1250) — compile-verified
//
#include <hip/hip_runtime.h>
#include <math.h>

// ---------------------------------------------------------------------------
// ChildSumTreeLSTM + attention over fixed memory, MI455X (gfx1250, wave32).
//
// Precompute (parallel, WMMA f32 16x16x4 + TDM-staged B panels):
//   Xiou[256,3072] = inputs @ W_ioux + b_ioux + b_iouh
//   Xf  [256,1024] = inputs @ W_fx   + b_fx   + b_fh
//   HW2 [512,1024] = H @ W_attnh[1024:,:] + b_attnh
//   colsumH[1024]  = sum_m H[m,:]
// Per step (sequential, L2-resident GEMVs, ~33 MB/step << 192 MB L2):
//   v    = h @ [W_iouh | W_fh]
//   gates: c,hnew
//   hv   = hnew @ W_attnh[:1024,:]
//   score_m = sum_c tanh(hv[c]+HW2[m,c]) * Wa[c] ; s = softmax(score)
//   h_att = colsumH - s@H + hnew  -> output row, next h
// ---------------------------------------------------------------------------

typedef __attribute__((ext_vector_type(2))) float        v2f;
typedef __attribute__((ext_vector_type(8))) float        v8f;
typedef __attribute__((ext_vector_type(4))) unsigned int u32x4;
typedef __attribute__((ext_vector_type(8))) int          i32x8;
typedef __attribute__((ext_vector_type(4))) int          i32x4;

#define T_STEPS 256
#define MEM     1024
#define MROWS   512

__device__ __forceinline__ float sigmoidf_(float x) {
    return 1.0f / (1.0f + __expf(-x));
}

// ---------------------------------------------------------------------------
// WMMA f32 GEMM with TDM B-panel staging.
// C[M,N] = A[M,K] @ B[K,N] + bias0 + bias1 (row-major, strides compile-time).
// Block = 256 threads (8 waves) -> 16(M) x 128(N) C tile; K chunked by 32.
// B chunk (32x128 f32 = 16 KB) is DMA'd to LDS by the Tensor Data Mover
// (one issue per chunk, wave 0 only; TDM ignores EXEC so guard per-wave),
// then all 8 waves feed WMMA B fragments from LDS.
// A frag (16x4 f32, ISA 7.12.2): lanes 0-15 (M=lane, K=0,1),
// lanes 16-31 (M=lane-16, K=2,3). C frag: VGPR j = rows j / j+8.
// ---------------------------------------------------------------------------
template<int LDA, int LDB, int LDC>
__global__ void wmma_gemm_f32_tdm(const float* __restrict__ A,
                                  const float* __restrict__ B,
                                  const float* __restrict__ bias0,
                                  const float* __restrict__ bias1,
                                  float* __restrict__ C, int K)
{
    __shared__ float ldsB[32 * 128];            // 16 KB B staging tile

    const int tid  = threadIdx.x;
    const int lane = tid & 31;
    const int wave = tid >> 5;
    const int n0b  = blockIdx.x * 128;          // block N origin
    const int m0   = blockIdx.y * 16;           // block M origin
    const int half = lane >> 4;                 // 0: K=0,1   1: K=2,3
    const int idx  = lane & 15;
    const int colL = wave * 16 + idx;           // column inside LDS tile

    if (tid == 0) ldsB[0] = 0.0f;               // keep LDS alias analysis honest

    const float* Arow = A + (size_t)(m0 + idx) * LDA + half * 2;

    v8f acc = {};

    for (int k0 = 0; k0 < K; k0 += 32) {
        __syncthreads();                        // prior tile fully consumed
        if (wave == 0) {
            // ---- Tensor DMA: B[k0:k0+32, n0b:n0b+128] -> ldsB ----
            unsigned long long ga =
                (unsigned long long)(const void*)(B + (size_t)k0 * LDB + n0b);
            unsigned lds = (unsigned)(size_t)(void*)ldsB;
            u32x4 g0;
            g0[0] = 1u;                                          // count=1
            g0[1] = lds;                                         // lds_addr
            g0[2] = (unsigned)ga;                                // gaddr[31:0]
            g0[3] = ((unsigned)(ga >> 32) & 0x01FFFFFFu)
                  | (2u << 30);                                  // type=2
            i32x8 g1;
            g1[0] = 2 << 16;                                     // data_size=4B
            g1[1] = (LDB & 0xFFFF) << 16;                        // tdim0 lo16
            g1[2] = (LDB >> 16) | ((K & 0xFFFF) << 16);          // hi | tdim1 lo
            g1[3] = ((unsigned)K >> 16) | (128u << 16);          // hi | tile0=128
            g1[4] = 32;                                          // tile1=32
            g1[5] = LDB;                                         // dim0_stride
            g1[6] = 0;
            g1[7] = 0;
            i32x4 z4 = {0, 0, 0, 0};                             // groups 2/3 NULL
            i32x8 z8 = {0, 0, 0, 0, 0, 0, 0, 0};
            __builtin_amdgcn_tensor_load_to_lds(g0, g1, z4, z4, z8, 0);
            __builtin_amdgcn_s_wait_tensorcnt(0);                // data in LDS
        }
        __syncthreads();                        // tile visible to all waves

        #pragma unroll
        for (int kl = 0; kl < 32; kl += 4) {
            v2f a = *(const v2f*)(Arow + k0 + kl);       // 8B-aligned
            v2f b;
            b[0] = ldsB[(kl + half * 2) * 128 + colL];
            b[1] = ldsB[(kl + half * 2 + 1) * 128 + colL];
            acc = __builtin_amdgcn_wmma_f32_16x16x4_f32(
                /*neg_a=*/false, a, /*neg_b=*/false, b,
                /*c_mod=*/(short)0, acc, /*reuse_a=*/false, /*reuse_b=*/false);
        }
    }

    const int cn = n0b + colL;
    float bs = 0.0f;
    if (bias0) bs += bias0[cn];
    if (bias1) bs += bias1[cn];
    const int mb = m0 + half * 8;
    #pragma unroll
    for (int j = 0; j < 8; ++j)
        C[(size_t)(mb + j) * LDC + cn] = acc[j] + bs;
}

// ---------------------------------------------------------------------------
// GEMV: y[n] = sum_k x[k] * W[k,n]  (W row-major [K,ldw]); K % 32 == 0.
// 256 threads = 64 columns x 4-way K split; coalesced 256B rows; L2 prefetch.
// ---------------------------------------------------------------------------
__global__ void gemv_kernel(const float* __restrict__ x,
                            const float* __restrict__ W, int ldw, int K,
                            float* __restrict__ y)
{
    __shared__ float red[256];
    const int tid = threadIdx.x;
    const int n   = blockIdx.x * 64 + (tid & 63);
    const int kk  = tid >> 6;                         // 0..3
    float acc = 0.0f;
    for (int kb = 0; kb < K; kb += 32) {
        if (kb + 64 < K)
            __builtin_prefetch(&W[(size_t)(kb + 64 + kk) * ldw + n], 0, 1);
        #pragma unroll
        for (int j = 0; j < 8; ++j) {
            const int k = kb + kk + 4 * j;
            acc += x[k] * W[(size_t)k * ldw + n];
        }
    }
    red[tid] = acc;
    __syncthreads();
    if (kk == 0)
        y[n] = red[tid] + red[tid + 64] + red[tid + 128] + red[tid + 192];
}

// gates: i,o,u from Xiou_t + v[0:3072], f from Xf_t + v[3072:4096]
__global__ void gates_kernel(const float* __restrict__ Xiou_t,
                             const float* __restrict__ Xf_t,
                             const float* __restrict__ v,
                             float* __restrict__ c,
                             float* __restrict__ hnew)
{
    const int n = blockIdx.x * blockDim.x + threadIdx.x;   // 0..1023
    const float iv = sigmoidf_(Xiou_t[n]        + v[n]);
    const float ov = sigmoidf_(Xiou_t[n + 1024] + v[n + 1024]);
    const float uv = tanhf    (Xiou_t[n + 2048] + v[n + 2048]);
    const float fv = sigmoidf_(Xf_t[n]          + v[3072 + n]);
    const float cn = iv * uv + fv * c[n];
    c[n]    = cn;
    hnew[n] = ov * tanhf(cn);
}

// scores[m] = sum_c tanh(hv[c] + HW2[m,c]) * Wa[c] ; one wave32 per row
__global__ void scores_kernel(const float* __restrict__ hv,
                              const float* __restrict__ HW2,
                              const float* __restrict__ Wa,
                              float* __restrict__ scores)
{
    const int lane = threadIdx.x & 31;
    const int wave = threadIdx.x >> 5;
    const int m    = blockIdx.x * 8 + wave;               // 0..511
    const float* row = HW2 + (size_t)m * MEM;
    float acc = 0.0f;
    for (int col = lane; col < MEM; col += 32)
        acc += tanhf(hv[col] + row[col]) * Wa[col];
    #pragma unroll
    for (int o = 16; o > 0; o >>= 1)
        acc += __shfl_xor(acc, o, 32);
    if (lane == 0) scores[m] = acc;
}

// softmax over 512 scores, single block of 512 threads
__global__ void softmax_kernel(const float* __restrict__ scores,
                               float* __restrict__ s)
{
    __shared__ float sh[512];
    const int tid = threadIdx.x;
    const float v = scores[tid];
    sh[tid] = v; __syncthreads();
    for (int o = 256; o > 0; o >>= 1) {
        if (tid < o) sh[tid] = fmaxf(sh[tid], sh[tid + o]);
        __syncthreads();
    }
    const float mx = sh[0];
    __syncthreads();
    const float e = __expf(v - mx);
    sh[tid] = e; __syncthreads();
    for (int o = 256; o > 0; o >>= 1) {
        if (tid < o) sh[tid] += sh[tid + o];
        __syncthreads();
    }
    s[tid] = e / sh[0];
}

// h_att[n] = colsumH[n] - (s @ H)[n] + hnew[n]; writes next-h and output row
__global__ void attn_out_kernel(const float* __restrict__ s,
                                const float* __restrict__ H,
                                const float* __restrict__ colsumH,
                                const float* __restrict__ hnew,
                                float* __restrict__ h,
                                float* __restrict__ out_row)
{
    __shared__ float red[256];
    const int tid = threadIdx.x;
    const int n   = blockIdx.x * 64 + (tid & 63);
    const int kk  = tid >> 6;
    float acc = 0.0f;
    for (int m = kk; m < MROWS; m += 4)
        acc += s[m] * H[(size_t)m * MEM + n];
    red[tid] = acc;
    __syncthreads();
    if (kk == 0) {
        const float tot = red[tid] + red[tid + 64] + red[tid + 128] + red[tid + 192];
        const float val = colsumH[n] - tot + hnew[n];
        h[n]       = val;
        out_row[n] = val;
    }
}

__global__ void colsum_kernel(const float* __restrict__ H,
                              float* __restrict__ cs)
{
    const int n = blockIdx.x * blockDim.x + threadIdx.x;
    float acc = 0.0f;
    for (int m = 0; m < MROWS; ++m) acc += H[(size_t)m * MEM + n];
    cs[n] = acc;
}

__global__ void zero_kernel(float* __restrict__ p)
{
    p[blockIdx.x * blockDim.x + threadIdx.x] = 0.0f;
}

extern "C" void kernel_launch(void* const* d_in, const int* in_sizes, int n_in,
                              void* d_out, int out_size, void* d_ws, size_t ws_size,
                              hipStream_t stream)
{
    const float* inputs  = (const float*)d_in[0];   // [256,1,1024]
    const float* H       = (const float*)d_in[1];   // [512,1024]
    const float* W_ioux  = (const float*)d_in[2];   // [1024,3072]
    const float* b_ioux  = (const float*)d_in[3];
    const float* W_iouh  = (const float*)d_in[4];   // [1024,3072]
    const float* b_iouh  = (const float*)d_in[5];
    const float* W_fx    = (const float*)d_in[6];   // [1024,1024]
    const float* b_fx    = (const float*)d_in[7];
    const float* W_fh    = (const float*)d_in[8];   // [1024,1024]
    const float* b_fh    = (const float*)d_in[9];
    const float* Wa      = (const float*)d_in[10];  // [1,1024]
    const float* W_attnh = (const float*)d_in[11];  // [2048,1024]
    const float* b_attnh = (const float*)d_in[12];
    float* out = (float*)d_out;                     // [256,1024]
    float* ws  = (float*)d_ws;

    // workspace layout (floats)
    float* Xiou = ws;                         // 256*3072
    float* Xf   = Xiou + 256 * 3072;          // 256*1024
    float* HW2  = Xf   + 256 * 1024;          // 512*1024
    float* cs   = HW2  + 512 * 1024;          // 1024
    float* h    = cs   + 1024;                // 1024 (c follows contiguously)
    float* c    = h    + 1024;                // 1024
    float* hnew = c    + 1024;                // 1024
    float* v    = hnew + 1024;                // 4096
    float* hv   = v    + 4096;                // 1024
    float* sc   = hv   + 1024;                // 512
    float* sm   = sc   + 512;                 // 512

    // ---- precompute (WMMA GEMMs with TDM-staged B panels) ----
    wmma_gemm_f32_tdm<1024, 3072, 3072><<<dim3(3072 / 128, 256 / 16), 256, 0, stream>>>(
        inputs, W_ioux, b_ioux, b_iouh, Xiou, 1024);
    wmma_gemm_f32_tdm<1024, 1024, 1024><<<dim3(1024 / 128, 256 / 16), 256, 0, stream>>>(
        inputs, W_fx, b_fx, b_fh, Xf, 1024);
    wmma_gemm_f32_tdm<1024, 1024, 1024><<<dim3(1024 / 128, 512 / 16), 256, 0, stream>>>(
        H, W_attnh + (size_t)1024 * 1024, b_attnh, nullptr, HW2, 1024);
    colsum_kernel<<<4, 256, 0, stream>>>(H, cs);
    zero_kernel<<<8, 256, 0, stream>>>(h);     // zeros h[1024] and c[1024]

    // ---- sequential scan (all weights stay L2-resident: ~40 MB << 192 MB) ----
    for (int t = 0; t < T_STEPS; ++t) {
        gemv_kernel<<<48, 256, 0, stream>>>(h, W_iouh, 3072, 1024, v);
        gemv_kernel<<<16, 256, 0, stream>>>(h, W_fh, 1024, 1024, v + 3072);
        gates_kernel<<<4, 256, 0, stream>>>(Xiou + (size_t)t * 3072,
                                            Xf + (size_t)t * 1024, v, c, hnew);
        gemv_kernel<<<16, 256, 0, stream>>>(hnew, W_attnh, 1024, 1024, hv);
        scores_kernel<<<64, 256, 0, stream>>>(hv, HW2, Wa, sc);
        softmax_kernel<<<1, 512, 0, stream>>>(sc, sm);
        attn_out_kernel<<<16, 256, 0, stream>>>(sm, H, cs, hnew, h,
                                                out + (size_t)t * 1024);
    }
}